// ReconStep_90082644066371
// MI455X (gfx1250) — compile-verified
//
#include <hip/hip_runtime.h>
#include <stdint.h>

#define GRID_N   128
#define LPB      8        // LORs per block, one wave32 each
#define BLOCK    256      // 8 waves

// Geometry constants (all three axes identical: 128^3 grid, 214^3 mm cube @ origin)
#define VOX      (214.0f / 128.0f)
#define INVVOX   (128.0f / 214.0f)
#define XMIN     (-107.0f)
// sigma^2 = vox^2/pi -> 1/(2 sigma^2) = pi/(2 vox^2); fold log2(e) for v_exp_f32 (exp2)
#define C2       ((3.14159265358979f / (2.0f * VOX * VOX)) * 1.44269504088896f)

#if __has_builtin(__builtin_amdgcn_exp2f)
#define EXP2F __builtin_amdgcn_exp2f
#else
#define EXP2F exp2f
#endif

typedef uint32_t v4u __attribute__((ext_vector_type(4)));
typedef int32_t  v8i __attribute__((ext_vector_type(8)));
typedef int32_t  v4i __attribute__((ext_vector_type(4)));

// DIR 0 (z-LORs): voxel (i,j,k) ; DIR 1 (x-LORs): (j,k,i) ; DIR 2 (y-LORs): (j,i,k)
template <int DIR>
__global__ __launch_bounds__(BLOCK) void project_bp_kernel(
    const float* __restrict__ img, const float* __restrict__ lors,
    float* __restrict__ bp, int nlor)
{
  constexpr int SI = (DIR == 0) ? 16384 : (DIR == 1) ? 1     : 128;
  constexpr int SJ = (DIR == 0) ? 128   : (DIR == 1) ? 16384 : 16384;
  constexpr int SK = (DIR == 0) ? 1     : (DIR == 1) ? 128   : 1;

  __shared__ float s_lor[6 * LPB];   // 6 components x 8 LORs

  // ---- TDM: DMA the 2D LOR tile (8 contiguous LORs x 6 rows strided by nlor)
  //      from global memory into LDS. Wave 0 issues; TENSORcnt tracks it.
  if (threadIdx.x < 32) {
    uint64_t gaddr = (uint64_t)(uintptr_t)(lors + (size_t)blockIdx.x * LPB);
    uint32_t ldsa  = (uint32_t)(uintptr_t)(&s_lor[0]);

    v4u g0; v8i g1; v4i g2 = {0, 0, 0, 0}; v4i g3 = {0, 0, 0, 0};
    // D# group 0: count=1 (valid), lds_addr, global_addr[56:0], type=2
    g0[0] = 1u;
    g0[1] = ldsa;
    g0[2] = (uint32_t)gaddr;
    g0[3] = ((uint32_t)(gaddr >> 32) & 0x01FFFFFFu) | (2u << 30);
    // D# group 1: data_size=4B, tensor_dim0=nlor, tensor_dim1=6,
    //             tile_dim0=8, tile_dim1=6, tile_dim2=0, dim0_stride=nlor
    uint32_t td0 = (uint32_t)nlor, td1 = 6u, t0 = (uint32_t)LPB, t1 = 6u;
    g1[0] = (int32_t)(2u << 16);                          // data_size=4B; mask=0
    g1[1] = (int32_t)((td0 & 0xFFFFu) << 16);             // abar_addr=0 | dim0.lo16
    g1[2] = (int32_t)((td0 >> 16) | ((td1 & 0xFFFFu) << 16));
    g1[3] = (int32_t)((td1 >> 16) | (t0 << 16));          // dim1.hi16 | tile_dim0
    g1[4] = (int32_t)(t1);                                // tile_dim1 | tile_dim2=0
    g1[5] = (int32_t)td0;                                 // dim0_stride lo32
    g1[6] = 0;
    g1[7] = 0;
#if defined(__clang_major__) && (__clang_major__ >= 23)
    v8i g4 = {0, 0, 0, 0, 0, 0, 0, 0};
    __builtin_amdgcn_tensor_load_to_lds(g0, g1, g2, g3, g4, 0);
#else
    __builtin_amdgcn_tensor_load_to_lds(g0, g1, g2, g3, 0);
#endif
    __builtin_amdgcn_s_wait_tensorcnt(0);
  }
  __syncthreads();

  const int wave = threadIdx.x >> 5;
  const int lane = threadIdx.x & 31;
  const int lor  = blockIdx.x * LPB + wave;
  if (lor >= nlor) return;

  const float p1x = s_lor[0 * LPB + wave];
  const float p1y = s_lor[1 * LPB + wave];
  const float p1z = s_lor[2 * LPB + wave];
  const float dx  = s_lor[3 * LPB + wave] - p1x;
  const float dy  = s_lor[4 * LPB + wave] - p1y;
  const float dzr = s_lor[5 * LPB + wave] - p1z;

  const float L     = sqrtf(dx * dx + dy * dy + dzr * dzr);
  const float dz    = (fabsf(dzr) < 1e-6f) ? 1e-6f : dzr;
  const float invdz = 1.0f / dz;
  const float dl    = VOX * L / fabsf(dz);

  // per-lane caches: 4 planes (k = lane + 32*m), separable masked 3-tap weights
  float wxc[4][3], wyc[4][3];
  int   i0c[4], j0c[4];
  unsigned tvm = 0;
  float s = 0.0f;

#pragma unroll
  for (int m = 0; m < 4; ++m) {
    const int   k  = lane + 32 * m;
    const float zc = XMIN + ((float)k + 0.5f) * VOX;
    const float t  = (zc - p1z) * invdz;
    const float xs = p1x + t * dx;
    const float ys = p1y + t * dy;
    const float fi = (xs - XMIN) * INVVOX - 0.5f;
    const float fj = (ys - XMIN) * INVVOX - 0.5f;
    const int   i0 = __float2int_rn(fi);   // round-to-nearest-even == jnp.round
    const int   j0 = __float2int_rn(fj);
    i0c[m] = i0; j0c[m] = j0;
    const bool tv = (t >= 0.0f) && (t <= 1.0f);
    if (tv) tvm |= (1u << m);

    int ro[3], co[3];
#pragma unroll
    for (int a = 0; a < 3; ++a) {
      // x taps: clip index (v_med3), zero weight when clipped (== reference)
      const int   ii  = i0 - 1 + a;
      const int   iic = min(max(ii, 0), GRID_N - 1);
      const float cx  = XMIN + ((float)ii + 0.5f) * VOX - xs;
      const float wx  = EXP2F(-(cx * cx) * C2);
      wxc[m][a] = (ii == iic) ? wx : 0.0f;
      ro[a]     = iic * SI;
      // y taps
      const int   jj  = j0 - 1 + a;
      const int   jjc = min(max(jj, 0), GRID_N - 1);
      const float cy  = XMIN + ((float)jj + 0.5f) * VOX - ys;
      const float wy  = EXP2F(-(cy * cy) * C2);
      wyc[m][a] = (jj == jjc) ? wy : 0.0f;
      co[a]     = jjc * SJ;
    }

    if (tv) {
      const int kt = k * SK;
#pragma unroll
      for (int a = 0; a < 3; ++a) {
        const float wa = wxc[m][a];
        const float* row = img + (ro[a] + kt);
#pragma unroll
        for (int b = 0; b < 3; ++b)
          s += wa * wyc[m][b] * row[co[b]];
      }
    }
  }

  // wave32 butterfly reduction: every lane ends with the full line sum
#pragma unroll
  for (int off = 16; off >= 1; off >>= 1) s += __shfl_xor(s, off, 32);

  const float proj  = s * dl;
  const float scale = dl / (proj + 1e-8f);

  // pass 2: scatter cached masked weights * scale with native fp32 atomics
#pragma unroll
  for (int m = 0; m < 4; ++m) {
    if (!(tvm & (1u << m))) continue;
    const int k  = lane + 32 * m;
    const int i0 = i0c[m], j0 = j0c[m];
    const int kt = k * SK;
#pragma unroll
    for (int a = 0; a < 3; ++a) {
      const int iic = min(max(i0 - 1 + a, 0), GRID_N - 1);
      const float wa = wxc[m][a] * scale;
      float* row = bp + (iic * SI + kt);
#pragma unroll
      for (int b = 0; b < 3; ++b) {
        const int jjc = min(max(j0 - 1 + b, 0), GRID_N - 1);
        unsafeAtomicAdd(row + jjc * SJ, wa * wyc[m][b]);
      }
    }
  }
}

__global__ __launch_bounds__(256) void finalize_kernel(
    const float4* __restrict__ img, const float4* __restrict__ eff,
    float4* out, int n4)
{
  const int idx = blockIdx.x * blockDim.x + threadIdx.x;
  if (idx < n4) {
    const float4 a = img[idx];
    const float4 e = eff[idx];
    float4 o = out[idx];
    o.x = a.x / (e.x + 1e-8f) * o.x;
    o.y = a.y / (e.y + 1e-8f) * o.y;
    o.z = a.z / (e.z + 1e-8f) * o.z;
    o.w = a.w / (e.w + 1e-8f) * o.w;
    out[idx] = o;
  }
}

extern "C" void kernel_launch(void* const* d_in, const int* in_sizes, int n_in,
                              void* d_out, int out_size, void* d_ws, size_t ws_size,
                              hipStream_t stream) {
  const float* img = (const float*)d_in[0];
  const float* eff = (const float*)d_in[1];
  const float* xl  = (const float*)d_in[2];
  const float* yl  = (const float*)d_in[3];
  const float* zl  = (const float*)d_in[4];
  float* out = (float*)d_out;

  const int nlor = in_sizes[2] / 6;   // lors stored as [6, N]

  // accumulate all three backprojections directly into d_out
  hipMemsetAsync(out, 0, (size_t)out_size * sizeof(float), stream);

  const int nblk = (nlor + LPB - 1) / LPB;
  project_bp_kernel<0><<<nblk, BLOCK, 0, stream>>>(img, zl, out, nlor);
  project_bp_kernel<1><<<nblk, BLOCK, 0, stream>>>(img, xl, out, nlor);
  project_bp_kernel<2><<<nblk, BLOCK, 0, stream>>>(img, yl, out, nlor);

  const int n4 = out_size / 4;
  finalize_kernel<<<(n4 + 255) / 256, 256, 0, stream>>>(
      (const float4*)img, (const float4*)eff, (float4*)out, n4);
}